// MarchingTetrahedrons_1047972021054
// MI455X (gfx1250) — compile-verified
//
#include <hip/hip_runtime.h>
#include <hip/hip_bf16.h>
#include <cstdint>
#include <cmath>

#define EMPTY_KEY 0xFFFFFFFFFFFFFFFFull
#define CHUNK 256

typedef __attribute__((ext_vector_type(4))) unsigned int v4u;
typedef __attribute__((ext_vector_type(4))) int          v4i;
typedef __attribute__((ext_vector_type(8))) int          v8i;

// ---- marching-tets tables (flattened) ----
__constant__ int d_TRI[96] = {
  -1,-1,-1,-1,-1,-1,  1,0,2,-1,-1,-1,  4,0,3,-1,-1,-1,
   1,4,2, 1, 3, 4,    3,1,5,-1,-1,-1,  2,3,0, 2, 5, 3,
   1,4,0, 1, 5, 4,    4,2,5,-1,-1,-1,  4,5,2,-1,-1,-1,
   4,1,0, 4, 5, 1,    3,2,0, 3, 5, 2,  1,3,5,-1,-1,-1,
   4,1,2, 4, 3, 1,    3,0,4,-1,-1,-1,  2,0,1,-1,-1,-1,
  -1,-1,-1,-1,-1,-1 };
__constant__ int d_NTRI[16] = {0,1,1,2,1,2,2,1,1,2,2,1,2,1,1,0};

// ---- CDNA5 helpers -------------------------------------------------------
// Generic pointers to LDS carry the DS byte offset in their low 32 bits
// (ISA 10.2, LDS aperture).
__device__ __forceinline__ unsigned lds_addr_of(const void* p) {
  return (unsigned)(uintptr_t)p;
}

// Async global -> LDS copies, GV addressing (64-bit vaddr, saddr=off).
__device__ __forceinline__ void async_load_b128_to_lds(unsigned lds,
                                                       unsigned long long gaddr) {
  asm volatile("global_load_async_to_lds_b128 %0, %1, off"
               :: "v"(lds), "v"(gaddr) : "memory");
}
__device__ __forceinline__ void async_load_b64_to_lds(unsigned lds,
                                                      unsigned long long gaddr) {
  asm volatile("global_load_async_to_lds_b64 %0, %1, off"
               :: "v"(lds), "v"(gaddr) : "memory");
}

__device__ __forceinline__ void wait_asynccnt0() {
  asm volatile("s_wait_asynccnt 0x0" ::: "memory");
}

__device__ __forceinline__ void wait_tensorcnt0() {
#if __has_builtin(__builtin_amdgcn_s_wait_tensorcnt)
  __builtin_amdgcn_s_wait_tensorcnt(0);
#else
  asm volatile("s_wait_tensorcnt 0x0" ::: "memory");
#endif
}

#if __has_builtin(__builtin_amdgcn_tensor_load_to_lds)
#define HAVE_TDM 1
#else
#define HAVE_TDM 0
#endif

#if HAVE_TDM
// TDM: 1-D tile of `tile_dwords` 4-byte elements, tensor_dim0 bounds the read
// (OOB reads return zero).  D# groups per CDNA5 ISA.
// The builtin keeps the descriptor in SGPR tuples for us (inline-asm "s"
// constraints cannot express an SReg_256, which broke last round).
__device__ __forceinline__ void tdm_load_1d_dwords(unsigned lds_byte,
                                                   unsigned long long gaddr,
                                                   unsigned tile_dwords,
                                                   unsigned long long tensor_dwords) {
  unsigned dim0 = tensor_dwords > 0xFFFFFFFFull ? 0xFFFFFFFFu
                                                : (unsigned)tensor_dwords;
  v4u g0;
  g0.x = 1u;                                              // count=1, user mode
  g0.y = lds_byte;                                        // lds_addr
  g0.z = (unsigned)gaddr;                                 // global_addr[31:0]
  g0.w = (unsigned)((gaddr >> 32) & 0x01FFFFFFull) | (2u << 30); // [56:32]|type=2
  v8i g1;
  g1.s0 = (int)(2u << 16);                    // wg_mask=0, data_size=2 (4 bytes)
  g1.s1 = (int)((dim0 & 0xFFFFu) << 16);      // tensor_dim0[15:0]  @ bits 63:48
  g1.s2 = (int)((dim0 >> 16) | (1u << 16));   // tensor_dim0[31:16] | tensor_dim1=1
  g1.s3 = (int)((tile_dwords & 0xFFFFu) << 16); // tensor_dim1 hi=0 | tile_dim0
  g1.s4 = 1;                                  // tile_dim1=1, tile_dim2=0
  g1.s5 = (int)dim0;                          // tensor_dim0_stride[31:0]
  g1.s6 = 0;                                  // stride0 hi | stride1 lo
  g1.s7 = 0;
  v4i z4 = {0, 0, 0, 0};
#if __has_include(<hip/amd_detail/amd_gfx1250_TDM.h>)
  // amdgpu-toolchain (clang-23 / therock-10.0): 6-arg form
  v8i z8 = {0, 0, 0, 0, 0, 0, 0, 0};
  __builtin_amdgcn_tensor_load_to_lds(g0, g1, z4, z4, z8, 0);
#else
  // ROCm 7.2 (clang-22): 5-arg form
  __builtin_amdgcn_tensor_load_to_lds(g0, g1, z4, z4, 0);
#endif
}
#endif // HAVE_TDM

// ---- open-addressing edge hash ------------------------------------------
__device__ __forceinline__ unsigned edge_hash(unsigned long long key, unsigned mask) {
  return (unsigned)((key * 0x9E3779B97F4A7C15ull) >> 32) & mask;
}

__device__ __forceinline__ void hash_insert(unsigned long long* keys,
                                            unsigned long long key, unsigned mask) {
  unsigned h = edge_hash(key, mask);
  for (unsigned i = 0; i <= mask; ++i) {
    unsigned long long prev = atomicCAS(&keys[h], EMPTY_KEY, key);
    if (prev == EMPTY_KEY || prev == key) return;
    h = (h + 1) & mask;
  }
}

__device__ __forceinline__ int hash_lookup(const unsigned long long* __restrict__ keys,
                                           const int* __restrict__ vals,
                                           unsigned long long key, unsigned mask) {
  unsigned h = edge_hash(key, mask);
  for (unsigned i = 0; i <= mask; ++i) {
    unsigned long long k = keys[h];
    if (k == key) return vals[h];
    if (k == EMPTY_KEY) return -1;
    h = (h + 1) & mask;
  }
  return -1;
}

// ---- K0: reset counters + hash keys (re-run every launch: determinism) ---
__global__ void k_init(int* __restrict__ counters,
                       unsigned long long* __restrict__ keys, size_t cap) {
  size_t i = (size_t)blockIdx.x * blockDim.x + threadIdx.x;
  if (i < 8) counters[i] = 0;
  for (; i < cap; i += (size_t)gridDim.x * blockDim.x) keys[i] = EMPTY_KEY;
}

// ---- K1: classify tets, compact valid list, insert crossing edges --------
__global__ void k_classify(const float* __restrict__ sdf,
                           const int4* __restrict__ tets, int num_tets,
                           int* __restrict__ counters,
                           int2* __restrict__ valid_list, int valid_cap,
                           unsigned long long* __restrict__ keys,
                           unsigned mask, int use_hash) {
  __shared__ int4 s_tet[CHUNK];
  const int tid  = threadIdx.x;
  const int base = blockIdx.x * CHUNK;
  const int gi   = base + tid;

  // Stage this block's tet indices through the async LDS pipe (ASYNCcnt).
  if (gi < num_tets)
    async_load_b128_to_lds(lds_addr_of(&s_tet[tid]),
                           (unsigned long long)(uintptr_t)(tets + gi));
  // Warm L2 for the next chunk (global_prefetch_b8).
  if (base + CHUNK + tid < num_tets)
    __builtin_prefetch(tets + base + CHUNK + tid, 0, 1);
  wait_asynccnt0();
  __syncthreads();
  if (gi >= num_tets) return;

  const int4 t = s_tet[tid];
  const int vv[4] = {t.x, t.y, t.z, t.w};
  int occ = 0;
  occ |= (sdf[vv[0]] > 0.f) ? 1 : 0;
  occ |= (sdf[vv[1]] > 0.f) ? 2 : 0;
  occ |= (sdf[vv[2]] > 0.f) ? 4 : 0;
  occ |= (sdf[vv[3]] > 0.f) ? 8 : 0;
  if (d_NTRI[occ] == 0) return;                 // occ_sum == 0 or 4

  int slot = atomicAdd(&counters[2], 1);
  if (slot < valid_cap) valid_list[slot] = make_int2(gi, occ);

  if (!use_hash) return;
#pragma unroll
  for (int e = 0; e < 6; ++e) {                 // BASE_TET_EDGES packed in nibbles
    int ia = (0x211000 >> (4 * e)) & 0xF;
    int ib = (0x332321 >> (4 * e)) & 0xF;
    if (((occ >> ia) & 1) != ((occ >> ib) & 1)) {
      unsigned a = (unsigned)vv[ia], b = (unsigned)vv[ib];
      unsigned lo = min(a, b), hi = max(a, b);
      hash_insert(keys, ((unsigned long long)lo << 32) | hi, mask);
    }
  }
}

// ---- K2: assign vertex ids to unique edges + interpolate -----------------
__global__ void k_assign(const float* __restrict__ pos,
                         const float* __restrict__ sdf,
                         unsigned long long* __restrict__ keys,
                         int* __restrict__ vals, unsigned cap,
                         int* __restrict__ counters,
                         float* __restrict__ verts, int verts_cap_elems) {
  unsigned s = blockIdx.x * blockDim.x + threadIdx.x;
  if (s >= cap) return;
  unsigned long long key = keys[s];
  if (key == EMPTY_KEY) return;
  int id = atomicAdd(&counters[0], 1);
  vals[s] = id;
  unsigned a = (unsigned)(key >> 32), b = (unsigned)key;
  float sa = sdf[a], sb = sdf[b];
  float denom = sa - sb;                         // (s_a) + (-s_b)
  float wa = -sb / denom, wb = sa / denom;       // flip / sum per reference
  int o = 3 * id;
  if (o + 2 < verts_cap_elems) {
    verts[o + 0] = pos[3 * a + 0] * wa + pos[3 * b + 0] * wb;
    verts[o + 1] = pos[3 * a + 1] * wa + pos[3 * b + 1] * wb;
    verts[o + 2] = pos[3 * a + 2] * wa + pos[3 * b + 2] * wb;
  }
}

// ---- K3: emit faces + uv indices (valid list streamed via TDM) -----------
__global__ void k_faces(const int2* __restrict__ valid_list, int valid_cap,
                        const unsigned long long* __restrict__ keys,
                        const int* __restrict__ vals, unsigned mask,
                        const int4* __restrict__ tets,
                        int* __restrict__ counters,
                        int* __restrict__ faces, int faces_cap_elems,
                        int* __restrict__ uvidx, int uvidx_cap_elems, int N) {
  __shared__ int2 s_valid[CHUNK];
  __shared__ int  s_tab[96];
  __shared__ int  s_nt[16];
  if (threadIdx.x < 96) s_tab[threadIdx.x] = d_TRI[threadIdx.x];
  if (threadIdx.x < 16) s_nt[threadIdx.x]  = d_NTRI[threadIdx.x];

#if HAVE_TDM
  // Wave 0 DMAs this block's slice of the compacted list via the TDM.
  if ((threadIdx.x >> 5) == 0) {
    unsigned long long gbase = (unsigned long long)(uintptr_t)valid_list
                             + (unsigned long long)blockIdx.x * (CHUNK * 8ull);
    unsigned long long total_dw = (unsigned long long)valid_cap * 2ull
                                - (unsigned long long)blockIdx.x * (CHUNK * 2ull);
    tdm_load_1d_dwords(lds_addr_of(&s_valid[0]), gbase, CHUNK * 2u, total_dw);
    wait_tensorcnt0();
  }
#else
  // Fallback: per-lane async b64 copies (still the CDNA5 ASYNCcnt pipe).
  {
    int idx = blockIdx.x * CHUNK + threadIdx.x;
    if (idx < valid_cap)
      async_load_b64_to_lds(lds_addr_of(&s_valid[threadIdx.x]),
                            (unsigned long long)(uintptr_t)(valid_list + idx));
    wait_asynccnt0();
  }
#endif
  __syncthreads();

  int nvalid = counters[2];
  if (nvalid > valid_cap) nvalid = valid_cap;
  int i = blockIdx.x * CHUNK + threadIdx.x;
  if (i >= nvalid) return;

  int2 vt  = s_valid[threadIdx.x];
  int  gi  = vt.x;
  int  occ = vt.y;
  const int4 t = tets[gi];
  const int vv[4] = {t.x, t.y, t.z, t.w};

  int ntri = s_nt[occ & 15];
  for (int tr = 0; tr < ntri; ++tr) {
    int fslot = atomicAdd(&counters[1], 1);
    int ids[3];
#pragma unroll
    for (int j = 0; j < 3; ++j) {
      int e  = s_tab[(occ & 15) * 6 + tr * 3 + j];
      int ia = (0x211000 >> (4 * e)) & 0xF;
      int ib = (0x332321 >> (4 * e)) & 0xF;
      unsigned a = (unsigned)vv[ia], b = (unsigned)vv[ib];
      unsigned lo = min(a, b), hi = max(a, b);
      ids[j] = hash_lookup(keys, vals,
                           ((unsigned long long)lo << 32) | hi, mask);
    }
    int o = 3 * fslot;
    if (o + 2 < faces_cap_elems) {
      faces[o + 0] = ids[0]; faces[o + 1] = ids[1]; faces[o + 2] = ids[2];
    }
    // uv_idx for face_gidx = gi*2 + tr
    long long fg = (long long)gi * 2 + tr;
    long long tet_idx = fg >> 1;
    int tri = (int)(fg & 1);
    if (o + 2 < uvidx_cap_elems) {
      uvidx[o + 0] = (int)(tet_idx * 4);
      uvidx[o + 1] = (int)(tet_idx * 4 + tri + 1);
      uvidx[o + 2] = (int)(tet_idx * 4 + tri + 2);
    }
  }
}

// ---- K4: uv grid (pure compute) ------------------------------------------
__global__ void k_uvs(float* __restrict__ uvs, int N, long long total_pairs,
                      long long cap_elems) {
  long long i = (long long)blockIdx.x * blockDim.x + threadIdx.x;
  if (i >= total_pairs) return;
  long long tcell = i >> 2;
  int c  = (int)(i & 3);
  int tx = (int)(tcell % N);
  int ty = (int)(tcell / N);
  float inv = 1.0f / (float)N;
  float pad = 0.9f / (float)N;
  float x = tx * inv + ((c == 1 || c == 2) ? pad : 0.0f);
  float y = ty * inv + ((c >= 2) ? pad : 0.0f);
  if (2 * i + 1 < cap_elems) {
    uvs[2 * i + 0] = x;
    uvs[2 * i + 1] = y;
  }
}

// ---- host ----------------------------------------------------------------
extern "C" void kernel_launch(void* const* d_in, const int* in_sizes, int n_in,
                              void* d_out, int out_size, void* d_ws, size_t ws_size,
                              hipStream_t stream) {
  const float* pos = (const float*)d_in[0];
  const float* sdf = (const float*)d_in[1];
  const int4*  tets = (const int4*)d_in[2];
  const int num_tets = in_sizes[2] / 4;

  // workspace partition: [counters | compacted valid list | hash keys | hash vals]
  char* ws = (char*)d_ws;
  int* counters = (int*)ws;
  size_t off = 256;
  size_t valid_cap = (size_t)num_tets;
  if (off + valid_cap * sizeof(int2) > ws_size)
    valid_cap = (ws_size > off) ? (ws_size - off) / sizeof(int2) : 0;
  int2* valid_list = (int2*)(ws + off);
  off += valid_cap * sizeof(int2);

  size_t remain = (ws_size > off) ? ws_size - off : 0;
  size_t cap = 0;
  if (remain >= 12 * 1024) {
    cap = 1024;
    while (cap * 2 * 12 <= remain) cap <<= 1;
  }
  unsigned long long* keys = (unsigned long long*)(ws + off);
  int* vals = (int*)(ws + off + cap * 8);

  // output partition: quarters of d_out (bounds-checked device side)
  float* outf = (float*)d_out;
  size_t q = (size_t)(out_size > 0 ? out_size : 0) / 4;
  float* o_verts = outf;
  int*   o_faces = (int*)(outf + q);
  float* o_uvs   = outf + 2 * q;
  int*   o_uvidx = (int*)(outf + 3 * q);

  k_init<<<2048, 256, 0, stream>>>(counters, keys, cap);

  int blocks = (num_tets + CHUNK - 1) / CHUNK;
  if (blocks > 0)
    k_classify<<<blocks, CHUNK, 0, stream>>>(
        sdf, tets, num_tets, counters, valid_list, (int)valid_cap,
        keys, (unsigned)(cap ? cap - 1 : 0), cap != 0);

  if (cap) {
    int ablocks = (int)((cap + CHUNK - 1) / CHUNK);
    k_assign<<<ablocks, CHUNK, 0, stream>>>(
        pos, sdf, keys, vals, (unsigned)cap, counters, o_verts, (int)q);
  }

  long long tet2 = (long long)num_tets * 2;
  int N = (int)ceil(sqrt((double)((tet2 + 1) / 2)));
  if (N < 1) N = 1;

  if (cap && valid_cap) {
    int fblocks = (int)((valid_cap + CHUNK - 1) / CHUNK);
    k_faces<<<fblocks, CHUNK, 0, stream>>>(
        valid_list, (int)valid_cap, keys, vals, (unsigned)(cap - 1),
        tets, counters, o_faces, (int)q, o_uvidx, (int)q, N);
  }

  long long pairs = (long long)N * N * 4;
  int ublocks = (int)((pairs + CHUNK - 1) / CHUNK);
  k_uvs<<<ublocks, CHUNK, 0, stream>>>(o_uvs, N, pairs, (long long)q);
}